// HBVMulTDET_WaterLoss_39419209843036
// MI455X (gfx1250) — compile-verified
//
#include <hip/hip_runtime.h>
#include <math.h>

#define LENF 15
#define NPAR 14     // only 14 of the 16 raw params are used by the model
#define PFD  16     // prefetch distance (timesteps) ~ HBM latency / iter latency

// ---- native CDNA5 transcendentals (v_log_f32 / v_exp_f32 / v_rcp_f32) ----
__device__ __forceinline__ float fast_log2(float x) { return __builtin_amdgcn_logf(x); }
__device__ __forceinline__ float fast_exp2(float x) { return __builtin_amdgcn_exp2f(x); }
__device__ __forceinline__ float fast_rcp (float x) { return __builtin_amdgcn_rcpf(x); }
// x > 0 guaranteed (states clamped to >= 1e-5)
__device__ __forceinline__ float fast_pow (float x, float e) { return fast_exp2(e * fast_log2(x)); }
__device__ __forceinline__ float clamp01  (float x) { return fminf(fmaxf(x, 0.f), 1.f); }

// sum over the 4 lanes of a quad (ensemble members) via ds_swizzle_b32
// group-of-32 encoding: offset = {xor[14:10], or[9:5], and[4:0]}
__device__ __forceinline__ float quad_sum(float v) {
  v += __int_as_float(__builtin_amdgcn_ds_swizzle(__float_as_int(v), 0x041F)); // XOR 1
  v += __int_as_float(__builtin_amdgcn_ds_swizzle(__float_as_int(v), 0x081F)); // XOR 2
  return v;
}

struct HBVState { float SP, MW, SM, SUZ, SLZ; };

// One HBV timestep; returns this member's Q0+Q1+Q2. Op order matches reference.
__device__ __forceinline__ float hbv_step(HBVState& s, const float cp[NPAR],
                                          float P, float Ta, float PET) {
  const float BETA   = fmaf(cp[0],    5.0f,   1.0f);
  const float FC     = fmaf(cp[1],  950.0f,  50.0f);
  const float K0     = fmaf(cp[2],   0.85f,  0.05f);
  const float K1     = fmaf(cp[3],   0.49f,  0.01f);
  const float K2     = fmaf(cp[4],  0.199f, 0.001f);
  const float LP     = fmaf(cp[5],    0.8f,   0.2f);
  const float PERC   = cp[6]  * 10.0f;
  const float UZL    = cp[7]  * 100.0f;
  const float TTp    = fmaf(cp[8],    5.0f,  -2.5f);
  const float CFMAX  = fmaf(cp[9],    9.5f,   0.5f);
  const float CFR    = cp[10] * 0.1f;
  const float CWH    = cp[11] * 0.2f;
  const float BETAET = fmaf(cp[12],   4.7f,   0.3f);
  const float Cc     = cp[13];

  const float rain = (Ta >= TTp) ? P : 0.f;
  const float snow = (Ta <  TTp) ? P : 0.f;
  s.SP += snow;
  const float melt = fminf(fmaxf(CFMAX * (Ta - TTp), 0.f), s.SP);
  s.MW += melt;
  s.SP = fmaxf(s.SP - melt, 1e-5f);
  const float refreeze = fminf(fmaxf(CFR * CFMAX * (TTp - Ta), 0.f), s.MW);
  s.SP += refreeze;
  s.MW = fmaxf(s.MW - refreeze, 1e-5f);
  const float tosoil = fmaxf(s.MW - CWH * s.SP, 0.f);
  s.MW = fmaxf(s.MW - tosoil, 1e-5f);

  const float rcpFC = fast_rcp(FC);
  const float soil_wet = clamp01(fast_pow(s.SM * rcpFC, BETA));
  const float recharge = (rain + tosoil) * soil_wet;
  s.SM = s.SM + rain + tosoil - recharge;
  const float excess = fmaxf(s.SM - FC, 0.f);
  s.SM = fmaxf(s.SM - excess, 1e-5f);
  const float evapfactor = clamp01(fast_pow(s.SM * rcpFC * fast_rcp(LP), BETAET));
  const float ETact = fminf(s.SM, PET * evapfactor);
  s.SM = fmaxf(s.SM - ETact, 1e-5f);
  const float capillary = fminf(s.SLZ, Cc * s.SLZ * (1.f - fminf(s.SM * rcpFC, 1.f)));
  s.SM += capillary;
  s.SLZ = fmaxf(s.SLZ - capillary, 1e-5f);

  s.SUZ += recharge + excess;
  const float PERCa = fminf(s.SUZ, PERC);
  s.SUZ -= PERCa;
  const float Q0 = K0 * fmaxf(s.SUZ - UZL, 0.f);
  s.SUZ -= Q0;
  const float Q1 = K1 * s.SUZ;
  s.SUZ -= Q1;
  s.SLZ += PERCa;
  const float Q2 = K2 * s.SLZ;
  s.SLZ -= Q2;
  return Q0 + Q1 + Q2;
}

__global__ __launch_bounds__(64, 1)
void hbv_fused_kernel(const float* __restrict__ xf,    // (T, N, 3)  P, Temp, PET
                      const float* __restrict__ praw,  // (T, N, 16, 4)
                      const float* __restrict__ conv,  // (N, 2)
                      float*       __restrict__ out,   // (T, N)
                      int N, int T)
{
  const int tid = blockIdx.x * blockDim.x + threadIdx.x;
  const int n = tid >> 2;   // basin
  const int m = tid & 3;    // ensemble member (quad lane)
  if (n >= N) return;

  // ---------------- gamma unit hydrograph (per basin) ----------------
  // exp(-lgamma(aa))/theta^aa is constant over k and cancels under w/w.sum().
  const float aa = fmaxf(conv[2 * n + 0] * 2.9f, 0.f) + 0.1f;
  const float th = fmaxf(conv[2 * n + 1] * 6.5f, 0.f) + 0.5f;
  const float e1 = aa - 1.0f;
  const float nls = -1.44269504088896f / th;  // exp(-tt/th) = exp2(nls*tt)
  float uh[LENF];
  float wsum = 0.f;
#pragma unroll
  for (int k = 0; k < LENF; ++k) {
    const float tt = (float)k + 0.5f;
    const float w = fast_exp2(e1 * fast_log2(tt) + nls * tt);
    uh[k] = w; wsum += w;
  }
  const float inv_wsum = 1.0f / wsum;
#pragma unroll
  for (int k = 0; k < LENF; ++k) uh[k] *= inv_wsum;

  // ---------------- state & routing ring (all registers) ----------------
  HBVState st = {0.001f, 0.001f, 0.001f, 0.001f, 0.001f};
  float hist[LENF - 1];
#pragma unroll
  for (int k = 0; k < LENF - 1; ++k) hist[k] = 0.f;

  // ---------------- streaming pointers ----------------
  const size_t pstride = (size_t)N * 64;  // elements per timestep in praw
  const size_t fstride = (size_t)N * 3;
  const float* pp = praw + (size_t)n * 64 + m;  // lane's param column, t = 0
  const float* fp = xf + (size_t)n * 3;
  // lane-parity offset so each quad prefetches BOTH 128B lines of the 256B row
  const int pline = (m & 1) * 32;               // +128 bytes in float units

  // ---- warm the prefetch pipeline: global_prefetch_b8 for first PFD steps ----
#pragma unroll
  for (int d = 1; d <= PFD; ++d) {
    if (d < T) {
      __builtin_prefetch(pp + (size_t)d * pstride + pline, 0, 1);
      __builtin_prefetch(fp + (size_t)d * fstride, 0, 1);
    }
  }

  // prologue: load t = 0 into "current" registers
  float cp[NPAR];
#pragma unroll
  for (int p = 0; p < NPAR; ++p) cp[p] = pp[p * 4];
  float cP = fp[0], cT = fp[1], cE = fp[2];

  for (int t = 0; t < T - 1; ++t) {
    // ---- issue next-step loads early (register double-buffer) ----
    const float* pn = pp + pstride;
    const float* fn = fp + fstride;
    float np_[NPAR];
#pragma unroll
    for (int p = 0; p < NPAR; ++p) np_[p] = pn[p * 4];
    const float nP = fn[0], nT = fn[1], nE = fn[2];

    if (t + 1 + PFD < T) {  // HBM-distance prefetch -> global_prefetch_b8
      __builtin_prefetch(pn + (size_t)PFD * pstride + pline, 0, 1);
      __builtin_prefetch(fn + (size_t)PFD * fstride, 0, 1);
    }

    // ---- step + ensemble mean + fused 15-tap UH routing ----
    const float q = hbv_step(st, cp, cP, cT, cE);
    const float qt = 0.25f * quad_sum(q);
    float qr = uh[0] * qt;
#pragma unroll
    for (int k = 1; k < LENF; ++k) qr = fmaf(uh[k], hist[k - 1], qr);
#pragma unroll
    for (int k = LENF - 2; k > 0; --k) hist[k] = hist[k - 1];
    hist[0] = qt;
    if (m == 0) out[(size_t)t * N + n] = qr;

    // ---- rotate double-buffer ----
    pp = pn; fp = fn;
#pragma unroll
    for (int p = 0; p < NPAR; ++p) cp[p] = np_[p];
    cP = nP; cT = nT; cE = nE;
  }

  // ---- peeled final step (no next-step loads) ----
  {
    const int t = T - 1;
    const float q = hbv_step(st, cp, cP, cT, cE);
    const float qt = 0.25f * quad_sum(q);
    float qr = uh[0] * qt;
#pragma unroll
    for (int k = 1; k < LENF; ++k) qr = fmaf(uh[k], hist[k - 1], qr);
    if (m == 0) out[(size_t)t * N + n] = qr;
  }
}

extern "C" void kernel_launch(void* const* d_in, const int* in_sizes, int n_in,
                              void* d_out, int out_size, void* d_ws, size_t ws_size,
                              hipStream_t stream) {
  const float* xf   = (const float*)d_in[0];  // (T, N, 3)
  const float* praw = (const float*)d_in[1];  // (T, N, 16, 4)
  const float* conv = (const float*)d_in[2];  // (N, 2)
  float* out = (float*)d_out;                 // (T, N)

  const int N = in_sizes[2] / 2;
  const int T = in_sizes[0] / (N * 3);

  const int threads = N * 4;                  // one thread per (basin, member)
  const int block = 64;                       // 16 basins x 4 members = 2 wave32
  const int grid = (threads + block - 1) / block;   // ~250 blocks -> ~250 WGPs
  hipLaunchKernelGGL(hbv_fused_kernel, dim3(grid), dim3(block), 0, stream,
                     xf, praw, conv, out, N, T);
}